// SigMax_21852793602397
// MI455X (gfx1250) — compile-verified
//
#include <hip/hip_runtime.h>
#include <hip/hip_bf16.h>
#include <math.h>

// ---------------------------------------------------------------------------
// SigMax pooling for MI455X (gfx1250, wave32).
//   K1: score = sigmoid(feat . W + b)              -> d_out score region
//   K2: per-batch valley segmentation (LDS scan)   -> ws seg tables + hlens_new
//   K3: banded GEMM feat_new = W^T * F via V_WMMA_F32_16X16X4_F32
// ---------------------------------------------------------------------------

typedef float v2f __attribute__((ext_vector_type(2)));
typedef float v8f __attribute__((ext_vector_type(8)));

// ---------------------------------------------------------------------------
// K1: one wave per (b,t) row; 512-float dot product, shfl reduction.
// ---------------------------------------------------------------------------
__global__ __launch_bounds__(256) void sigmax_score_kernel(
    const float4* __restrict__ feat4,   // [rows, D/4]
    const float4* __restrict__ W4,      // [D/4]
    const float*  __restrict__ bias,    // [1]
    float* __restrict__ score,          // [rows]
    int rows, int D4)
{
    int wave = threadIdx.x >> 5;
    int lane = threadIdx.x & 31;
    int row  = blockIdx.x * 8 + wave;
    if (row >= rows) return;

    const float4* fr = feat4 + (size_t)row * D4;
    float acc = 0.f;
    for (int j = lane; j < D4; j += 32) {
        float4 f = fr[j];
        float4 w = W4[j];
        acc += f.x * w.x + f.y * w.y + f.z * w.z + f.w * w.w;
    }
    #pragma unroll
    for (int off = 16; off; off >>= 1)
        acc += __shfl_xor(acc, off, 32);

    if (lane == 0)
        score[row] = 1.0f / (1.0f + __expf(-(acc + bias[0])));
}

// ---------------------------------------------------------------------------
// K2: one wave per batch. Stage score row in LDS, run a wave-uniform serial
// scan to find valleys (local maxima, forced at t=0 and t=hl-1), build
// per-segment [vstart, vend) and 1/(segsum+1e-10) tables, emit hlens_new.
// Segment k covers [vi[k-1], vi[k+1]) per the reference slicing.
// ---------------------------------------------------------------------------
__global__ __launch_bounds__(32) void sigmax_segment_kernel(
    const float* __restrict__ score,    // [B,T] (in d_out)
    const int*   __restrict__ hlens,    // [B]
    int*   __restrict__ hlens_new,      // [B] (in d_out, int bits)
    int*   __restrict__ vstart,         // [B,Mpad] (ws)
    int*   __restrict__ vend,           // [B,Mpad] (ws)
    float* __restrict__ invsum,         // [B,Mpad] (ws)
    int T, int Mpad)
{
    __shared__ float s_sc[2048];
    __shared__ int   s_vi[2004];
    __shared__ float s_cs[2004];

    int b    = blockIdx.x;
    int lane = threadIdx.x;
    const float* sc = score + (size_t)b * T;

    for (int t = lane; t < T; t += 32) s_sc[t] = sc[t];
    __syncthreads();

    int   hl     = hlens[b];
    int   c      = 0;       // valley count (uniform across lanes)
    float run    = 0.f;     // exclusive prefix sum of score
    float cs_end = 0.f;     // csum at t = hl-1

    for (int t = 0; t < hl; ++t) {
        float s = s_sc[t];
        float l = (t > 0)     ? s_sc[t - 1] : s;
        float r = (t + 1 < T) ? s_sc[t + 1] : s;
        bool valley = (t == 0) || (t == hl - 1) || ((s >= l) && (s >= r));
        if (t == hl - 1) cs_end = run;
        if (valley) {
            if (lane == 0 && c < 2004) { s_vi[c] = t; s_cs[c] = run; }
            ++c;
        }
        run += s;
    }
    __syncthreads();

    if (lane == 0) hlens_new[b] = c;

    size_t base = (size_t)b * Mpad;
    for (int k = lane; k < Mpad; k += 32) {
        int vs = 0, ve = 0;
        float iv = 0.f;
        if (k < c) {
            vs        = (k == 0) ? 0   : s_vi[k - 1];
            float cs0 = (k == 0) ? 0.f : s_cs[k - 1];
            float cs1;
            if (k == c - 1) { ve = hl - 1;      cs1 = cs_end;      }
            else            { ve = s_vi[k + 1]; cs1 = s_cs[k + 1]; }
            iv = 1.0f / (cs1 - cs0 + 1e-10f);
        }
        vstart[base + k] = vs;
        vend  [base + k] = ve;
        invsum[base + k] = iv;
    }
}

// ---------------------------------------------------------------------------
// K3: banded GEMM with V_WMMA_F32_16X16X4_F32.
// Block = 8 waves; block owns (batch b, 16-segment tile k0..k0+15).
// Wave w owns d-range [64w, 64w+64): 4 accumulator tiles of 16x16 f32.
// A tile (16 segs x 4 t): lanes 0-15 hold K=0,1; lanes 16-31 hold K=2,3
// (documented 32-bit A layout). Weights built in-register from seg tables.
// B tile (4 t x 16 d): feat rows, mirrored half-lane K split.
// t-loop walks only [vstart[k0], max(vend) ) -- the tile's live band.
// TD = compile-time D (512 fast path) so store offsets fold to immediates.
// ---------------------------------------------------------------------------
template<int TD>
__global__ __launch_bounds__(256) void sigmax_wmma_kernel(
    const float* __restrict__ feat,     // [B,T,D]
    const float* __restrict__ score,    // [B,T]
    const int*   __restrict__ vstart,   // [B,Mpad]
    const int*   __restrict__ vend,     // [B,Mpad]
    const float* __restrict__ invsum,   // [B,Mpad]
    float* __restrict__ out,            // [B,M,D]
    int T, int Drt, int M, int Mpad, int numTiles)
{
    const int D = TD ? TD : Drt;

    int tile = blockIdx.x % numTiles;
    int b    = blockIdx.x / numTiles;
    int k0   = tile * 16;
    int lane = threadIdx.x & 31;
    int wave = threadIdx.x >> 5;

    size_t sb  = (size_t)b * Mpad;
    int seg    = k0 + (lane & 15);
    int vs     = vstart[sb + seg];
    int ve     = vend  [sb + seg];
    float iv   = invsum[sb + seg];

    // uniform loop bounds for the wave: band covered by these 16 segments
    int tstart = __shfl(vs, 0, 32);
    int tmax   = ve;
    #pragma unroll
    for (int off = 16; off; off >>= 1)
        tmax = max(tmax, __shfl_xor(tmax, off, 32));

    const float* sc = score + (size_t)b * T;
    const float* fb = feat  + (size_t)b * T * D;
    int d0 = wave * 64;
    int dl = d0 + (lane & 15);

    v8f acc0 = {}, acc1 = {}, acc2 = {}, acc3 = {};

    int thalf = (lane >= 16) ? 2 : 0;   // A/B half-lane K split

    for (int base = tstart; base < tmax; base += 4) {
        int ta  = base + thalf;
        int tb  = ta + 1;
        int tca = min(ta, T - 1);
        int tcb = min(tb, T - 1);

        // A tile: per-lane (t, seg) weights, zero outside [vstart, vend)
        float sa  = sc[tca];
        float sbv = sc[tcb];
        v2f a;
        a.x = (ta >= vs && ta < ve) ? sa  * iv : 0.f;
        a.y = (tb >= vs && tb < ve) ? sbv * iv : 0.f;

        // B tiles: feat rows tca/tcb, 4 x 16 columns
        const float* ra = fb + (size_t)tca * D;
        const float* rb = fb + (size_t)tcb * D;
        v2f b0; b0.x = ra[dl];      b0.y = rb[dl];
        v2f b1; b1.x = ra[dl + 16]; b1.y = rb[dl + 16];
        v2f b2; b2.x = ra[dl + 32]; b2.y = rb[dl + 32];
        v2f b3; b3.x = ra[dl + 48]; b3.y = rb[dl + 48];

        // prefetch next chunk's row (global_prefetch_b8, speculative)
        __builtin_prefetch(ra + 4 * (size_t)D + dl, 0, 1);

        acc0 = __builtin_amdgcn_wmma_f32_16x16x4_f32(false, a, false, b0,
                                                     (short)0, acc0, false, false);
        acc1 = __builtin_amdgcn_wmma_f32_16x16x4_f32(false, a, false, b1,
                                                     (short)0, acc1, false, false);
        acc2 = __builtin_amdgcn_wmma_f32_16x16x4_f32(false, a, false, b2,
                                                     (short)0, acc2, false, false);
        acc3 = __builtin_amdgcn_wmma_f32_16x16x4_f32(false, a, false, b3,
                                                     (short)0, acc3, false, false);
    }

    // C/D layout: VGPR r -> row r (lanes 0-15) / row r+8 (lanes 16-31)
    int rowHalf  = (lane >= 16) ? 8 : 0;
    int drel     = lane & 15;
    int segRow0  = k0 + rowHalf;
    float* po    = out + ((size_t)b * M + segRow0) * D + d0 + drel;

    if (k0 + 16 <= M) {
        // full tile: single base address, constant byte offsets (TD known)
        #pragma unroll
        for (int r = 0; r < 8; ++r) {
            po[(size_t)r * D +  0] = acc0[r];
            po[(size_t)r * D + 16] = acc1[r];
            po[(size_t)r * D + 32] = acc2[r];
            po[(size_t)r * D + 48] = acc3[r];
        }
    } else {
        // edge tile: guard rows beyond M
        #pragma unroll
        for (int r = 0; r < 8; ++r) {
            if (segRow0 + r < M) {
                po[(size_t)r * D +  0] = acc0[r];
                po[(size_t)r * D + 16] = acc1[r];
                po[(size_t)r * D + 32] = acc2[r];
                po[(size_t)r * D + 48] = acc3[r];
            }
        }
    }
}

// ---------------------------------------------------------------------------
// Launch
// ---------------------------------------------------------------------------
extern "C" void kernel_launch(void* const* d_in, const int* in_sizes, int n_in,
                              void* d_out, int out_size, void* d_ws, size_t ws_size,
                              hipStream_t stream) {
    const float* feat  = (const float*)d_in[0];  // [B,T,D]
    const float* W     = (const float*)d_in[1];  // [D]
    const float* bias  = (const float*)d_in[2];  // [1]
    const int*   hlens = (const int*)  d_in[3];  // [B]

    const int B = in_sizes[3];
    const int D = in_sizes[1];
    const int T = in_sizes[0] / (B * D);

    // out = feat_new [B,M,D] ++ hlens_new [B] ++ score [B,T]
    const int M        = (out_size - B - B * T) / (B * D);
    const int Mpad     = ((M + 15) / 16) * 16;
    const int numTiles = Mpad / 16;

    float* outf      = (float*)d_out;
    float* score     = outf + (size_t)B * M * D + B;   // [B,T]
    int*   hlens_new = (int*)(outf + (size_t)B * M * D);

    // workspace: vstart | vend | invsum, each [B,Mpad]
    int*   vstart = (int*)d_ws;
    int*   vend   = vstart + (size_t)B * Mpad;
    float* invsum = (float*)(vend + (size_t)B * Mpad);

    // K1: score
    int rows = B * T;
    sigmax_score_kernel<<<(rows + 7) / 8, 256, 0, stream>>>(
        (const float4*)feat, (const float4*)W, bias, score, rows, D / 4);

    // K2: segmentation
    sigmax_segment_kernel<<<B, 32, 0, stream>>>(
        score, hlens, hlens_new, vstart, vend, invsum, T, Mpad);

    // K3: banded WMMA GEMM (D=512 specialization folds store offsets)
    if (D == 512) {
        sigmax_wmma_kernel<512><<<B * numTiles, 256, 0, stream>>>(
            feat, score, vstart, vend, invsum, outf, T, D, M, Mpad, numTiles);
    } else {
        sigmax_wmma_kernel<0><<<B * numTiles, 256, 0, stream>>>(
            feat, score, vstart, vend, invsum, outf, T, D, M, Mpad, numTiles);
    }
}